// EDFFN_2688649527437
// MI455X (gfx1250) — compile-verified
//
#include <hip/hip_runtime.h>
#include <math.h>

// CDNA5 / gfx1250, wave32. D = A*B + C, fp32, 16x16x4.
typedef float v2f __attribute__((ext_vector_type(2)));
typedef float v8f __attribute__((ext_vector_type(8)));

__device__ __forceinline__ v8f wmma_f32(v2f a, v2f b, v8f c) {
  // 8-arg pattern: (neg_a, A, neg_b, B, c_mod, C, reuse_a, reuse_b)
  return __builtin_amdgcn_wmma_f32_16x16x4_f32(false, a, false, b, (short)0, c,
                                               false, false);
}

// ---------------------------------------------------------------------------
// Kernel P: turn the real rfft2 filter into a per-channel 8x8 circular-conv
// kernel.  y = irfft2(rfft2(p) * f)  ==  circconv(p, K_c) with
//   K[d1,d2] = (1/64) * sum_{a=0..7, b=0..4} w_b * f[a,b] * cos(2pi(a*d1+b*d2)/8)
//   w_b = 1 for b in {0,4} (DC/Nyquist columns: irfft takes real part), else 2.
// For the all-ones filter this is exactly the identity (delta kernel).
// ---------------------------------------------------------------------------
__global__ void kspec_kernel(const float* __restrict__ f, float* __restrict__ K) {
  int i = blockIdx.x * blockDim.x + threadIdx.x;   // 64*64 entries
  if (i >= 64 * 64) return;
  int c = i >> 6, d = i & 63, d1 = d >> 3, d2 = d & 7;
  const float PI4 = 0.78539816339744831f;          // 2*pi/8
  float s = 0.f;
  for (int a = 0; a < 8; ++a) {
    for (int b = 0; b < 5; ++b) {
      float w = (b == 0 || b == 4) ? 1.f : 2.f;
      int ph = (a * d1 + b * d2) & 7;
      s += w * f[c * 40 + a * 5 + b] * __cosf(PI4 * (float)ph);
    }
  }
  K[i] = s * (1.0f / 64.0f);
}

// ---------------------------------------------------------------------------
// Fully fused EDFFN: one workgroup per 8x8 output patch.
//   grid = (32*32, B), block = 256 threads (8 wave32).
// All intermediates live in LDS; only x is read and out is written to HBM
// (weights are tiny and L2-resident).
// ---------------------------------------------------------------------------
__global__ __launch_bounds__(256, 1)
void edffn_fused(const float* __restrict__ x, const float* __restrict__ w_in,
                 const float* __restrict__ b_in, const float* __restrict__ w_dw,
                 const float* __restrict__ b_dw, const float* __restrict__ w_out,
                 const float* __restrict__ b_out, const float* __restrict__ kspec,
                 float* __restrict__ out) {
  // LDS layout (strides padded for bank-conflict-free WMMA fragment loads)
  __shared__ float s_x[100 * 66];    // x halo: [pixel 0..99][cin 0..63], stride 66
  __shared__ float s_wdw[512 * 9];   // depthwise weights
  __shared__ float s_bdw[512];
  __shared__ float s_bin[512];
  __shared__ float s_bout[64];
  __shared__ float s_kf[64 * 65];    // circular-conv kernels, stride 65
  __shared__ float s_h[100 * 128];   // project_in chunk: [pixel][128 ch]; aliased as y
  __shared__ float s_g[64 * 258];    // gated activations: [pixel 0..63][256 ch], stride 258

  const int tid = threadIdx.x;
  const int b   = blockIdx.y;
  const int py  = blockIdx.x >> 5;
  const int pxw = blockIdx.x & 31;
  const int gy0 = py * 8 - 1;        // halo origin (row)
  const int gx0 = pxw * 8 - 1;       // halo origin (col)

  // ---- stage x halo (zeros outside image) + weights/biases/kernels ----
  for (int i = tid; i < 64 * 100; i += 256) {
    int ch = i / 100, p = i - ch * 100;
    int hr = p / 10, hc = p - hr * 10;
    int gy = gy0 + hr, gx = gx0 + hc;
    float v = 0.f;
    if ((unsigned)gy < 256u && (unsigned)gx < 256u)
      v = x[((b * 64 + ch) * 256 + gy) * 256 + gx];
    s_x[p * 66 + ch] = v;
  }
  for (int i = tid; i < 512 * 9; i += 256) s_wdw[i] = w_dw[i];
  for (int i = tid; i < 512; i += 256) { s_bdw[i] = b_dw[i]; s_bin[i] = b_in[i]; }
  if (tid < 64) s_bout[tid] = b_out[tid];
  for (int i = tid; i < 4096; i += 256) s_kf[(i >> 6) * 65 + (i & 63)] = kspec[i];
  __syncthreads();

  const int lane = tid & 31;
  const int wv   = tid >> 5;   // wave id 0..7
  const int col  = lane & 15;  // M-row (A) / N-col (B,D) index
  const int hi   = lane >> 4;  // selects k pair {2hi, 2hi+1} within each K=4 step

  // ---- 4 chunks: project_in GEMM (WMMA) -> dwconv 3x3 + GELU gate ----
  for (int cc = 0; cc < 4; ++cc) {
    if (cc) __syncthreads();   // previous dwconv must finish reading s_h

    // GEMM1: M = 112 (100 halo px, padded), N = 128 h-channels, K = 64.
    // Wave wv owns N-tile nt = wv; its h-channel (row of w_in):
    int nl  = wv * 16 + col;                            // 0..127
    int hch = (nl < 64) ? (cc * 64 + nl) : (192 + cc * 64 + nl);
    float bin_c = s_bin[hch];

    v2f bf[16];
#pragma unroll
    for (int ks = 0; ks < 16; ++ks) {
      const float* p = w_in + hch * 64 + ks * 4 + hi * 2;
      bf[ks] = (v2f){p[0], p[1]};
    }

    for (int mt = 0; mt < 7; ++mt) {
      v8f acc = {0.f, 0.f, 0.f, 0.f, 0.f, 0.f, 0.f, 0.f};
      const float* ap = &s_x[(mt * 16 + col) * 66 + hi * 2];
#pragma unroll
      for (int ks = 0; ks < 16; ++ks) {
        v2f a = (v2f){ap[ks * 4], ap[ks * 4 + 1]};
        acc = wmma_f32(a, bf[ks], acc);
      }
      // store h (+bias), zeroing SAME-padding halo pixels
#pragma unroll
      for (int v = 0; v < 8; ++v) {
        int pl = mt * 16 + v + 8 * hi;
        if (pl < 100) {
          int gy = gy0 + pl / 10, gx = gx0 + pl % 10;
          float val = ((unsigned)gy < 256u && (unsigned)gx < 256u)
                          ? (acc[v] + bin_c) : 0.f;
          s_h[pl * 128 + nl] = val;
        }
      }
    }
    __syncthreads();

    // depthwise 3x3 + exact GELU gate -> s_g (64 g-channels this chunk)
    for (int it = tid; it < 4096; it += 256) {
      int j = it & 63;           // channel within chunk
      int p = it >> 6;           // output pixel 0..63
      int u = p >> 3, vv = p & 7;
      int cl = cc * 64 + j;      // gate "x1" channel == g channel
      int chh = cl + 256;        // gate "x2" channel
      const float* wl = &s_wdw[cl * 9];
      const float* wh = &s_wdw[chh * 9];
      float alo = s_bdw[cl], ahi = s_bdw[chh];
#pragma unroll
      for (int dy = 0; dy < 3; ++dy)
#pragma unroll
        for (int dx = 0; dx < 3; ++dx) {
          int hp = (u + dy) * 10 + (vv + dx);
          alo += wl[dy * 3 + dx] * s_h[hp * 128 + j];
          ahi += wh[dy * 3 + dx] * s_h[hp * 128 + 64 + j];
        }
      float ge = 0.5f * alo * (1.0f + erff(alo * 0.70710678118654752f));
      s_g[p * 258 + cl] = ge * ahi;
    }
  }
  __syncthreads();

  // ---- GEMM2 (WMMA): y[64 px][64 ch] = g[64][256] @ w_out^T, + b_out ----
  float* s_y = s_h;  // alias (stride 66); safe after the barrier above
  for (int t = wv; t < 16; t += 8) {
    int mt = t >> 2, nt = t & 3;
    v8f acc = {0.f, 0.f, 0.f, 0.f, 0.f, 0.f, 0.f, 0.f};
    const float* ap = &s_g[(mt * 16 + col) * 258 + hi * 2];
    int n = nt * 16 + col;
    const float* bp = w_out + n * 256 + hi * 2;   // tiny, L2-resident
#pragma unroll
    for (int ks = 0; ks < 64; ++ks) {
      v2f a  = (v2f){ap[ks * 4], ap[ks * 4 + 1]};
      v2f bb = (v2f){bp[ks * 4], bp[ks * 4 + 1]};
      acc = wmma_f32(a, bb, acc);
    }
    float bo = s_bout[n];
#pragma unroll
    for (int v = 0; v < 8; ++v) {
      int pl = mt * 16 + v + 8 * hi;              // 0..63
      s_y[pl * 66 + n] = acc[v] + bo;
    }
  }
  __syncthreads();

  // ---- spectral filter: per-channel 8x8 circular convolution, store out ----
  for (int it = tid; it < 4096; it += 256) {
    int c = it >> 6;             // channel (uniform per wave -> broadcast reads)
    int p = it & 63;
    int u = p >> 3, vv = p & 7;
    const float* kf = &s_kf[c * 65];
    float s = 0.f;
#pragma unroll
    for (int sy = 0; sy < 8; ++sy)
#pragma unroll
      for (int tx = 0; tx < 8; ++tx)
        s += kf[(((u - sy) & 7) << 3) | ((vv - tx) & 7)] * s_y[(sy * 8 + tx) * 66 + c];
    int gy = py * 8 + u, gx = pxw * 8 + vv;
    out[((b * 64 + c) * 256 + gy) * 256 + gx] = s;
  }
}

extern "C" void kernel_launch(void* const* d_in, const int* in_sizes, int n_in,
                              void* d_out, int out_size, void* d_ws, size_t ws_size,
                              hipStream_t stream) {
  const float* x     = (const float*)d_in[0];
  const float* w_in  = (const float*)d_in[1];
  const float* b_in  = (const float*)d_in[2];
  const float* w_dw  = (const float*)d_in[3];
  const float* b_dw  = (const float*)d_in[4];
  const float* w_out = (const float*)d_in[5];
  const float* b_out = (const float*)d_in[6];
  const float* fftf  = (const float*)d_in[7];
  float* out   = (float*)d_out;
  float* kspec = (float*)d_ws;   // 64*64 floats = 16 KB

  kspec_kernel<<<16, 256, 0, stream>>>(fftf, kspec);
  dim3 grid(32 * 32, 4);
  edffn_fused<<<grid, 256, 0, stream>>>(x, w_in, b_in, w_dw, b_dw,
                                        w_out, b_out, kspec, out);
}